// PSA_65773129171431
// MI455X (gfx1250) — compile-verified
//
#include <hip/hip_runtime.h>
#include <math.h>

typedef float v2f __attribute__((ext_vector_type(2)));
typedef float v8f __attribute__((ext_vector_type(8)));

#define B_   16
#define C_   512
#define HW_  4096
#define IN_  256

__device__ __forceinline__ float wave_sum(float s) {
#pragma unroll
  for (int off = 16; off; off >>= 1) s += __shfl_xor(s, off, 32);
  return s;
}
__device__ __forceinline__ float wave_max(float s) {
#pragma unroll
  for (int off = 16; off; off >>= 1) s = fmaxf(s, __shfl_xor(s, off, 32));
  return s;
}
__device__ __forceinline__ float sigmoidf(float v) {
  return 1.0f / (1.0f + __expf(-v));
}

// ---------------------------------------------------------------------------
// K1: one streaming pass over x.
//   xsum_part[(b*4+nch)*512 + c] = sum over this block's 1024-n range of x[b,c,n]
//   cm_part[(b*4+cs)*4096 + n]   = sum over this block's 128-c range of Wcq[c]*x[b,c,n]
// grid = (nchunk=4, b=16, csplit=4), block = 256, each thread owns 4 n (float4)
// ---------------------------------------------------------------------------
__global__ __launch_bounds__(256) void k1_pass(const float* __restrict__ x,
                                               const float* __restrict__ Wcq,
                                               float* __restrict__ xsum_part,
                                               float* __restrict__ cm_part) {
  __shared__ float part[128 * 8];
  const int tid = threadIdx.x;
  const int wid = tid >> 5, lane = tid & 31;
  const int nch = blockIdx.x;     // 0..3  -> n range of 1024
  const int b   = blockIdx.y;     // 0..15
  const int cs  = blockIdx.z;     // 0..3  -> c range of 128
  const int n0  = nch * 1024 + tid * 4;
  const int c0  = cs * 128;
  const float4* xb = (const float4*)(x + ((size_t)b * C_ + c0) * HW_ + n0);

  float4 cm_acc = {0.f, 0.f, 0.f, 0.f};
  for (int ci = 0; ci < 128; ++ci) {
    float4 v = xb[(size_t)ci * (HW_ / 4)];
    float wq = Wcq[c0 + ci];                      // uniform -> scalar load
    cm_acc.x = fmaf(wq, v.x, cm_acc.x);
    cm_acc.y = fmaf(wq, v.y, cm_acc.y);
    cm_acc.z = fmaf(wq, v.z, cm_acc.z);
    cm_acc.w = fmaf(wq, v.w, cm_acc.w);
    float s = wave_sum((v.x + v.y) + (v.z + v.w));
    if (lane == 0) part[ci * 8 + wid] = s;
  }
  __syncthreads();
  if (tid < 128) {
    float s = 0.f;
#pragma unroll
    for (int j = 0; j < 8; ++j) s += part[tid * 8 + j];
    xsum_part[(b * 4 + nch) * C_ + c0 + tid] = s;
  }
  *(float4*)(cm_part + (size_t)(b * 4 + cs) * HW_ + n0) = cm_acc;
}

// ---------------------------------------------------------------------------
// K2a: reduce cm partials over csplit, softmax over HW per batch.
// grid = 16 (one per batch), block = 256, 16 n per thread (4x float4)
// ---------------------------------------------------------------------------
__global__ __launch_bounds__(256) void k2a_softmax(const float* __restrict__ cm_part,
                                                   float* __restrict__ cm) {
  __shared__ float red[8];
  const int b = blockIdx.x, tid = threadIdx.x;
  const int lane = tid & 31, wid = tid >> 5;

  float4 v[4];
  float lmax = -3.0e38f;
#pragma unroll
  for (int j = 0; j < 4; ++j) {
    int n = tid * 4 + j * 1024;
    float4 s = {0.f, 0.f, 0.f, 0.f};
#pragma unroll
    for (int p = 0; p < 4; ++p) {
      float4 q = *(const float4*)(cm_part + (size_t)(b * 4 + p) * HW_ + n);
      s.x += q.x; s.y += q.y; s.z += q.z; s.w += q.w;
    }
    v[j] = s;
    lmax = fmaxf(lmax, fmaxf(fmaxf(s.x, s.y), fmaxf(s.z, s.w)));
  }
  lmax = wave_max(lmax);
  if (lane == 0) red[wid] = lmax;
  __syncthreads();
  float m = red[0];
#pragma unroll
  for (int j = 1; j < 8; ++j) m = fmaxf(m, red[j]);
  __syncthreads();

  float lsum = 0.f;
#pragma unroll
  for (int j = 0; j < 4; ++j) {
    v[j].x = __expf(v[j].x - m); v[j].y = __expf(v[j].y - m);
    v[j].z = __expf(v[j].z - m); v[j].w = __expf(v[j].w - m);
    lsum += (v[j].x + v[j].y) + (v[j].z + v[j].w);
  }
  lsum = wave_sum(lsum);
  if (lane == 0) red[wid] = lsum;
  __syncthreads();
  float tot = 0.f;
#pragma unroll
  for (int j = 0; j < 8; ++j) tot += red[j];
  float inv = 1.0f / tot;
#pragma unroll
  for (int j = 0; j < 4; ++j) {
    int n = tid * 4 + j * 1024;
    float4 o = {v[j].x * inv, v[j].y * inv, v[j].z * inv, v[j].w * inv};
    *(float4*)(cm + (size_t)b * HW_ + n) = o;
  }
}

// ---------------------------------------------------------------------------
// K2b: spatial-branch small GEMMs via V_WMMA_F32_16X16X4_F32.
//   xmean (16x512) = reduce(xsum_part)/HW
//   avg_pre (16x256) = xmean @ Wsq^T     [WMMA, M=16 N=256 K=512]
//   avg = softmax(avg_pre, axis=inner)
//   w (16x512) = avg @ Wsv               [WMMA, M=16 N=512 K=256]
// 1 block, 512 threads = 16 waves; wave -> 16-wide N tile(s).
// A layout per ISA: lane<16 holds A[lr][k0..k0+1], lane>=16 holds A[lr][k0+2..3].
// B mirrors: vgpr0 = rows k0 / k0+2, vgpr1 = rows k0+1 / k0+3.
// ---------------------------------------------------------------------------
__global__ __launch_bounds__(512) void k2b_spatial(const float* __restrict__ xsum_part,
                                                   const float* __restrict__ Wsq,
                                                   const float* __restrict__ Wsv,
                                                   float* __restrict__ w_out) {
  __shared__ float smA[B_ * C_];    // xmean 32KB
  __shared__ float smS[B_ * IN_];   // avg   16KB
  const int tid = threadIdx.x;
  const int wid = tid >> 5, lane = tid & 31;
  const int half = lane >> 4, lr = lane & 15;

  for (int i = tid; i < B_ * C_; i += 512) {
    int b = i >> 9, c = i & (C_ - 1);
    float s = 0.f;
#pragma unroll
    for (int p = 0; p < 4; ++p) s += xsum_part[(b * 4 + p) * C_ + c];
    smA[i] = s * (1.0f / (float)HW_);
  }
  __syncthreads();

  { // GEMM1: avg_pre = xmean @ Wsq^T ; B[kk][n] = Wsq[n*512 + kk]
    v8f acc = {};
    const int n = (wid << 4) + lr;
    const float* Brow = Wsq + (size_t)n * C_;
    for (int k0 = 0; k0 < C_; k0 += 4) {
      v2f a;  a.x  = smA[(lr << 9) + k0 + 2 * half];
              a.y  = smA[(lr << 9) + k0 + 2 * half + 1];
      v2f bb; bb.x = Brow[k0 + 2 * half];
              bb.y = Brow[k0 + 2 * half + 1];
      acc = __builtin_amdgcn_wmma_f32_16x16x4_f32(false, a, false, bb,
                                                  (short)0, acc, false, false);
    }
#pragma unroll
    for (int r = 0; r < 8; ++r)
      smS[(r + 8 * half) * IN_ + (wid << 4) + lr] = acc[r];
  }
  __syncthreads();

  { // softmax over inner=256 : wave wid owns batch row wid
    float vals[8];
    float vmax = -3.0e38f;
#pragma unroll
    for (int j = 0; j < 8; ++j) {
      vals[j] = smS[(wid << 8) + lane + (j << 5)];
      vmax = fmaxf(vmax, vals[j]);
    }
    vmax = wave_max(vmax);
    float ssum = 0.f;
#pragma unroll
    for (int j = 0; j < 8; ++j) { vals[j] = __expf(vals[j] - vmax); ssum += vals[j]; }
    ssum = wave_sum(ssum);
    float inv = 1.0f / ssum;
#pragma unroll
    for (int j = 0; j < 8; ++j) smS[(wid << 8) + lane + (j << 5)] = vals[j] * inv;
  }
  __syncthreads();

  // GEMM2: w = avg @ Wsv ; B[kk][n] = Wsv[kk*512 + n] ; 2 tiles per wave
#pragma unroll
  for (int t = 0; t < 2; ++t) {
    const int n0 = (wid << 4) + t * 256;
    v8f acc = {};
    for (int k0 = 0; k0 < IN_; k0 += 4) {
      v2f a;  a.x  = smS[(lr << 8) + k0 + 2 * half];
              a.y  = smS[(lr << 8) + k0 + 2 * half + 1];
      v2f bb; bb.x = Wsv[(size_t)(k0 + 2 * half) * C_ + n0 + lr];
              bb.y = Wsv[(size_t)(k0 + 2 * half + 1) * C_ + n0 + lr];
      acc = __builtin_amdgcn_wmma_f32_16x16x4_f32(false, a, false, bb,
                                                  (short)0, acc, false, false);
    }
#pragma unroll
    for (int r = 0; r < 8; ++r)
      w_out[(r + 8 * half) * C_ + n0 + lr] = acc[r];
  }
}

// ---------------------------------------------------------------------------
// K3: second streaming pass over x (same grid/layout as K1).
//   xc_part[(b*4+nch)*512 + c] = sum over n-range of x[b,c,n]*cm[b,n]
//   ctx_part[(b*4+cs)*4096 + n] = sum over c-range of w[b,c]*x[b,c,n]
// ---------------------------------------------------------------------------
__global__ __launch_bounds__(256) void k3_pass(const float* __restrict__ x,
                                               const float* __restrict__ cm,
                                               const float* __restrict__ wrow,
                                               float* __restrict__ xc_part,
                                               float* __restrict__ ctx_part) {
  __shared__ float part[128 * 8];
  const int tid = threadIdx.x;
  const int wid = tid >> 5, lane = tid & 31;
  const int nch = blockIdx.x;
  const int b   = blockIdx.y;
  const int cs  = blockIdx.z;
  const int n0  = nch * 1024 + tid * 4;
  const int c0  = cs * 128;
  const float4* xb = (const float4*)(x + ((size_t)b * C_ + c0) * HW_ + n0);
  const float4 cmv = *(const float4*)(cm + (size_t)b * HW_ + n0);

  float4 ctx_acc = {0.f, 0.f, 0.f, 0.f};
  for (int ci = 0; ci < 128; ++ci) {
    float4 v = xb[(size_t)ci * (HW_ / 4)];
    float wv = wrow[b * C_ + c0 + ci];            // uniform -> scalar load
    ctx_acc.x = fmaf(wv, v.x, ctx_acc.x);
    ctx_acc.y = fmaf(wv, v.y, ctx_acc.y);
    ctx_acc.z = fmaf(wv, v.z, ctx_acc.z);
    ctx_acc.w = fmaf(wv, v.w, ctx_acc.w);
    float p = fmaf(v.x, cmv.x, fmaf(v.y, cmv.y, fmaf(v.z, cmv.z, v.w * cmv.w)));
    p = wave_sum(p);
    if (lane == 0) part[ci * 8 + wid] = p;
  }
  __syncthreads();
  if (tid < 128) {
    float s = 0.f;
#pragma unroll
    for (int j = 0; j < 8; ++j) s += part[tid * 8 + j];
    xc_part[(b * 4 + nch) * C_ + c0 + tid] = s;
  }
  *(float4*)(ctx_part + (size_t)(b * 4 + cs) * HW_ + n0) = ctx_acc;
}

// ---------------------------------------------------------------------------
// K4: channel-branch small GEMMs + LayerNorm + sigmoid -> mask (16x512).
//   xc (16x512) = reduce(xc_part)
//   context (16x256) = xc @ Wcv^T        [WMMA]
//   z (16x512) = context @ Wcz^T         [WMMA]
//   mask = sigmoid(LN(z)*gamma+beta)
// ---------------------------------------------------------------------------
__global__ __launch_bounds__(512) void k4_mask(const float* __restrict__ xc_part,
                                               const float* __restrict__ Wcv,
                                               const float* __restrict__ Wcz,
                                               const float* __restrict__ gamma,
                                               const float* __restrict__ beta,
                                               float* __restrict__ mask_out) {
  __shared__ float smX[B_ * C_];    // xc       32KB
  __shared__ float smK[B_ * IN_];   // context  16KB
  __shared__ float smZ[B_ * C_];    // z        32KB
  const int tid = threadIdx.x;
  const int wid = tid >> 5, lane = tid & 31;
  const int half = lane >> 4, lr = lane & 15;

  for (int i = tid; i < B_ * C_; i += 512) {
    int b = i >> 9, c = i & (C_ - 1);
    float s = 0.f;
#pragma unroll
    for (int p = 0; p < 4; ++p) s += xc_part[(b * 4 + p) * C_ + c];
    smX[i] = s;
  }
  __syncthreads();

  { // GEMM1: context = xc @ Wcv^T ; B[kk][n] = Wcv[n*512 + kk]
    v8f acc = {};
    const int n = (wid << 4) + lr;
    const float* Brow = Wcv + (size_t)n * C_;
    for (int k0 = 0; k0 < C_; k0 += 4) {
      v2f a;  a.x  = smX[(lr << 9) + k0 + 2 * half];
              a.y  = smX[(lr << 9) + k0 + 2 * half + 1];
      v2f bb; bb.x = Brow[k0 + 2 * half];
              bb.y = Brow[k0 + 2 * half + 1];
      acc = __builtin_amdgcn_wmma_f32_16x16x4_f32(false, a, false, bb,
                                                  (short)0, acc, false, false);
    }
#pragma unroll
    for (int r = 0; r < 8; ++r)
      smK[(r + 8 * half) * IN_ + (wid << 4) + lr] = acc[r];
  }
  __syncthreads();

  // GEMM2: z = context @ Wcz^T ; B[kk][n] = Wcz[n*256 + kk] ; 2 tiles per wave
#pragma unroll
  for (int t = 0; t < 2; ++t) {
    const int n = (wid << 4) + t * 256 + lr;
    const float* Brow = Wcz + (size_t)n * IN_;
    v8f acc = {};
    for (int k0 = 0; k0 < IN_; k0 += 4) {
      v2f a;  a.x  = smK[(lr << 8) + k0 + 2 * half];
              a.y  = smK[(lr << 8) + k0 + 2 * half + 1];
      v2f bb; bb.x = Brow[k0 + 2 * half];
              bb.y = Brow[k0 + 2 * half + 1];
      acc = __builtin_amdgcn_wmma_f32_16x16x4_f32(false, a, false, bb,
                                                  (short)0, acc, false, false);
    }
#pragma unroll
    for (int r = 0; r < 8; ++r)
      smZ[(r + 8 * half) * C_ + n] = acc[r];
  }
  __syncthreads();

  { // LayerNorm over C=512 + sigmoid; wave wid owns batch row wid
    float vals[16];
    float s = 0.f;
#pragma unroll
    for (int j = 0; j < 16; ++j) {
      vals[j] = smZ[(wid << 9) + lane + (j << 5)];
      s += vals[j];
    }
    s = wave_sum(s);
    float mu = s * (1.0f / (float)C_);
    float vs = 0.f;
#pragma unroll
    for (int j = 0; j < 16; ++j) { float d = vals[j] - mu; vs = fmaf(d, d, vs); }
    vs = wave_sum(vs);
    float rstd = rsqrtf(vs * (1.0f / (float)C_) + 1e-5f);
#pragma unroll
    for (int j = 0; j < 16; ++j) {
      int c = lane + (j << 5);
      float zn = (vals[j] - mu) * rstd * gamma[c] + beta[c];
      mask_out[(wid << 9) + c] = sigmoidf(zn);
    }
  }
}

// K4b: reduce ctx partials over csplit and apply sigmoid once. grid=64 x 256
__global__ __launch_bounds__(256) void k4b_sctx(const float* __restrict__ ctx_part,
                                                float* __restrict__ sctx) {
  const int idx = blockIdx.x * 256 + threadIdx.x;   // float4 index, B*HW/4
  const int b = idx >> 10, n4 = idx & 1023;
  float4 s = {0.f, 0.f, 0.f, 0.f};
#pragma unroll
  for (int p = 0; p < 4; ++p) {
    float4 q = ((const float4*)(ctx_part + (size_t)(b * 4 + p) * HW_))[n4];
    s.x += q.x; s.y += q.y; s.z += q.z; s.w += q.w;
  }
  float4 o = {sigmoidf(s.x), sigmoidf(s.y), sigmoidf(s.z), sigmoidf(s.w)};
  ((float4*)(sctx + (size_t)b * HW_))[n4] = o;
}

// K5: out = x * (mask[b,c] + sctx[b,n]).  grid = 32768 x 256, float4 per thread
__global__ __launch_bounds__(256) void k5_out(const float* __restrict__ x,
                                              const float* __restrict__ mask,
                                              const float* __restrict__ sctx,
                                              float* __restrict__ out) {
  const size_t e = ((size_t)blockIdx.x * 256 + threadIdx.x) * 4;
  const int b = (int)(e >> 21);                 // C_*HW_ = 2^21
  const int rem = (int)(e & ((1u << 21) - 1));
  const int c = rem >> 12;                      // HW_ = 2^12
  const int n = rem & (HW_ - 1);
  const float m = mask[(b << 9) + c];
  float4 xv = *(const float4*)(x + e);
  float4 sv = *(const float4*)(sctx + ((size_t)b << 12) + n);
  float4 o = {xv.x * (m + sv.x), xv.y * (m + sv.y),
              xv.z * (m + sv.z), xv.w * (m + sv.w)};
  *(float4*)(out + e) = o;
}

extern "C" void kernel_launch(void* const* d_in, const int* in_sizes, int n_in,
                              void* d_out, int out_size, void* d_ws, size_t ws_size,
                              hipStream_t stream) {
  (void)in_sizes; (void)n_in; (void)out_size; (void)ws_size;
  const float* x     = (const float*)d_in[0];
  const float* Wsq   = (const float*)d_in[1];
  const float* Wsv   = (const float*)d_in[2];
  const float* Wcq   = (const float*)d_in[3];
  const float* Wcv   = (const float*)d_in[4];
  const float* Wcz   = (const float*)d_in[5];
  const float* gamma = (const float*)d_in[6];
  const float* beta  = (const float*)d_in[7];
  float* out = (float*)d_out;
  float* ws  = (float*)d_ws;

  // workspace layout (floats); every slot is written before it is read
  float* xsum_part = ws;                       // 16*4*512   = 32768
  float* cm_part   = ws + 32768;               // 16*4*4096  = 262144
  float* cm        = ws + 294912;              // 16*4096    = 65536
  float* wrow      = ws + 360448;              // 16*512     = 8192
  float* xc_part   = ws + 368640;              // 16*4*512   = 32768
  float* ctx_part  = ws + 401408;              // 16*4*4096  = 262144
  float* sctx      = ws + 663552;              // 16*4096    = 65536
  float* maskb     = ws + 729088;              // 16*512     = 8192

  dim3 gpass(4, 16, 4);
  k1_pass    <<<gpass, 256, 0, stream>>>(x, Wcq, xsum_part, cm_part);
  k2a_softmax<<<16,    256, 0, stream>>>(cm_part, cm);
  k2b_spatial<<<1,     512, 0, stream>>>(xsum_part, Wsq, Wsv, wrow);
  k3_pass    <<<gpass, 256, 0, stream>>>(x, cm, wrow, xc_part, ctx_part);
  k4_mask    <<<1,     512, 0, stream>>>(xc_part, Wcv, Wcz, gamma, beta, maskb);
  k4b_sctx   <<<64,    256, 0, stream>>>(ctx_part, sctx);
  k5_out     <<<32768, 256, 0, stream>>>(x, maskb, sctx, out);
}